// MultiHeadAttention_86457691669080
// MI455X (gfx1250) — compile-verified
//
#include <hip/hip_runtime.h>
#include <math.h>

// ---------------------------------------------------------------------------
// MI455X (gfx1250) fused graph-attention forward.
// Precision: fp32 end-to-end via V_WMMA_F32_16X16X4_F32 (matches reference).
// Hot loops use uniform-SGPR-base + lane-invariant-voffset + imm addressing.
// ---------------------------------------------------------------------------

#define Bb 4
#define Nn 1024
#define Dd 1024
#define Hh 16
#define HDm 64
#define NORMF 0.03125f   // 1/sqrt(1024)

typedef __attribute__((ext_vector_type(2))) float v2f;
typedef __attribute__((ext_vector_type(8))) float v8f;

__device__ __forceinline__ v8f wmma4(v2f a, v2f b, v8f c) {
  // D = A(16x4,f32) * B(4x16,f32) + C(16x16,f32)
  return __builtin_amdgcn_wmma_f32_16x16x4_f32(
      /*neg_a=*/false, a, /*neg_b=*/false, b,
      /*c_mod=*/(short)0, c, /*reuse_a=*/false, /*reuse_b=*/false);
}

// ---------------------------------------------------------------------------
// C[m, j] = sum_k X[m,k] * W[j,k] * (SCALED ? kscale[batch,k] : 1) + bias[j]
// Wave: 16x128 C tile (8 accumulators). Workgroup (4 waves): 64x128.
// A and B operands are both contiguous float2 loads (row-major X, row-major W).
// ---------------------------------------------------------------------------
template <bool SCALED>
__global__ __launch_bounds__(128) void gemm_xwT_wmma(
    const float* __restrict__ X, const float* __restrict__ W,
    const float* __restrict__ bias, const float* __restrict__ kscale,
    float* __restrict__ Out, int Kd, int Nout, int rowsPerBatch)
{
  const int wave = threadIdx.x >> 5;
  const int lane = threadIdx.x & 31;
  const int col  = lane & 15;
  const int hi   = lane >> 4;
  const int mtile = blockIdx.x * 64 + wave * 16;
  const int nbase = blockIdx.y * 128;
  const int batch = (blockIdx.x * 64) / rowsPerBatch;

  // loop-invariant per-lane offsets (voffset); uniform pointers advance in SALU
  const int aoff = (mtile + col) * Kd + 2 * hi;
  int boff[8];
#pragma unroll
  for (int t = 0; t < 8; ++t) boff[t] = (nbase + t * 16 + col) * Kd + 2 * hi;
  const int soff = batch * Kd + 2 * hi;

  v8f acc[8];
#pragma unroll
  for (int t = 0; t < 8; ++t)
#pragma unroll
    for (int r = 0; r < 8; ++r) acc[t][r] = 0.0f;

  const float* xk = X;
  const float* wk = W;
  const float* sk = kscale;
  for (int k = 0; k < Kd; k += 4) {
    v2f a = *(const v2f*)(xk + aoff);
    if (SCALED) a = a * (*(const v2f*)(sk + soff));
#pragma unroll
    for (int t = 0; t < 8; ++t) {
      v2f bt = *(const v2f*)(wk + boff[t]);
      acc[t] = wmma4(a, bt, acc[t]);
    }
    xk += 4;
    wk += 4;
    if (SCALED) sk += 4;
  }

  float* orow = Out + (size_t)mtile * Nout + nbase;
  const int ooff = 8 * hi * Nout + col;   // + imm r*Nout + t*16
#pragma unroll
  for (int t = 0; t < 8; ++t) {
    const float bval = bias[nbase + t * 16 + col];
#pragma unroll
    for (int r = 0; r < 8; ++r)
      orow[ooff + r * Nout + t * 16] = acc[t][r] + bval;
  }
}

// ---------------------------------------------------------------------------
// Flash-style dual-softmax attention for one (b, h, 64-q-row block).
// Each wave owns a 16-row q tile. Never materializes the NxN score maps.
// Fuses: scores*scale + adj, global softmax, masked local softmax, both P@V,
// sigmoid gate mix, signed-sqrt. Writes tmp[B,N,DV].
// ---------------------------------------------------------------------------
__global__ __launch_bounds__(128) void attn_flash_dual(
    const float* __restrict__ Qp, const float* __restrict__ Kp,
    const float* __restrict__ Vp, const float* __restrict__ adj,
    const int* __restrict__ mask, const float* __restrict__ alpha,
    float* __restrict__ tmp)
{
  __shared__ float ldsP[4][2][256];   // per-wave C-layout -> A-layout staging

  const int wave = threadIdx.x >> 5;
  const int lane = threadIdx.x & 31;
  const int col  = lane & 15;
  const int hi   = lane >> 4;
  const int h    = blockIdx.y;
  const int b    = blockIdx.z;
  const int qb   = blockIdx.x * 64 + wave * 16;

  // Q tile (16 x 64) in WMMA A-format, pre-scaled by 1/sqrt(dk).
  v2f qa[16];
  {
    const float* qrow = Qp + (size_t)(b * Nn + qb + col) * Dd + h * HDm + 2 * hi;
#pragma unroll
    for (int c = 0; c < 16; ++c) {
      v2f q = *(const v2f*)(qrow + 4 * c);
      qa[c] = q * NORMF;
    }
  }

  v8f og[4], ol[4];
  float mg[8], lg[8], ml[8], ll[8];
#pragma unroll
  for (int t = 0; t < 4; ++t)
#pragma unroll
    for (int r = 0; r < 8; ++r) { og[t][r] = 0.0f; ol[t][r] = 0.0f; }
#pragma unroll
  for (int r = 0; r < 8; ++r) { mg[r] = -3.0e38f; ml[r] = -3.0e38f; lg[r] = 0.0f; ll[r] = 0.0f; }

  // uniform tile pointers (advance per k-tile in SALU) + lane-const voffsets
  const float* kt = Kp  + (size_t)b * Nn * Dd + h * HDm;
  const float* vt = Vp  + (size_t)b * Nn * Dd + h * HDm;
  const float* at = adj + (size_t)(b * Nn + qb) * Nn;
  const int*   mt = mask + (size_t)(b * Nn + qb) * Nn;
  const int koff   = col * Dd + 2 * hi;        // K B-operand lane offset
  const int adjoff = 8 * hi * Nn + col;        // adj/mask lane offset (+ imm r*Nn)
  const int voffD  = 2 * hi * Dd + col;        // V B-operand lane offset

  for (int j = 0; j < Nn / 16; ++j) {
    // ---- scores tile: S = (Q*scale) K^T  (16 WMMAs over hd=64) ----
    v8f s;
#pragma unroll
    for (int r = 0; r < 8; ++r) s[r] = 0.0f;
#pragma unroll
    for (int c = 0; c < 16; ++c) {
      v2f kv = *(const v2f*)(kt + koff + 4 * c);   // imm offset 16*c bytes
      s = wmma4(qa[c], kv, s);
    }
    if (j + 1 < Nn / 16)
      __builtin_prefetch(kt + 16 * Dd + koff, 0, 1);

    // ---- dual online softmax (C-layout rows live in 16-lane halves) ----
    v8f cgv, clv;
#pragma unroll
    for (int r = 0; r < 8; ++r) {
      const float sv = s[r] + at[adjoff + r * Nn];

      // global map
      float rm = sv;
      rm = fmaxf(rm, __shfl_xor(rm, 1, 16));
      rm = fmaxf(rm, __shfl_xor(rm, 2, 16));
      rm = fmaxf(rm, __shfl_xor(rm, 4, 16));
      rm = fmaxf(rm, __shfl_xor(rm, 8, 16));
      float mnew = fmaxf(mg[r], rm);
      float corr = __expf(mg[r] - mnew);
      float p    = __expf(sv - mnew);
      float ps = p;
      ps += __shfl_xor(ps, 1, 16);
      ps += __shfl_xor(ps, 2, 16);
      ps += __shfl_xor(ps, 4, 16);
      ps += __shfl_xor(ps, 8, 16);
      lg[r] = lg[r] * corr + ps;
      mg[r] = mnew;
      cgv[r] = corr;
      ldsP[wave][0][(r + 8 * hi) * 16 + col] = p;

      // local (masked) map
      const float sl = (mt[adjoff + r * Nn] == 0) ? -1.0e9f : sv;
      float rl = sl;
      rl = fmaxf(rl, __shfl_xor(rl, 1, 16));
      rl = fmaxf(rl, __shfl_xor(rl, 2, 16));
      rl = fmaxf(rl, __shfl_xor(rl, 4, 16));
      rl = fmaxf(rl, __shfl_xor(rl, 8, 16));
      float mlew = fmaxf(ml[r], rl);
      float corl = __expf(ml[r] - mlew);
      float pl   = __expf(sl - mlew);
      float pls = pl;
      pls += __shfl_xor(pls, 1, 16);
      pls += __shfl_xor(pls, 2, 16);
      pls += __shfl_xor(pls, 4, 16);
      pls += __shfl_xor(pls, 8, 16);
      ll[r] = ll[r] * corl + pls;
      ml[r] = mlew;
      clv[r] = corl;
      ldsP[wave][1][(r + 8 * hi) * 16 + col] = pl;
    }

    // rescale running accumulators (per-row correction, broadcast in halves)
#pragma unroll
    for (int t = 0; t < 4; ++t) { og[t] = og[t] * cgv; ol[t] = ol[t] * clv; }

    __builtin_amdgcn_wave_barrier();

    // ---- P @ V: one V operand pair feeds BOTH softmax accumulators ----
#pragma unroll
    for (int c = 0; c < 4; ++c) {
      v2f pga = *(const v2f*)&ldsP[wave][0][col * 16 + 4 * c + 2 * hi];
      v2f pla = *(const v2f*)&ldsP[wave][1][col * 16 + 4 * c + 2 * hi];
#pragma unroll
      for (int t = 0; t < 4; ++t) {
        v2f bv;
        bv.x = vt[voffD + 4 * c * Dd + t * 16];        // imm 16384*c + 64*t B
        bv.y = vt[voffD + 4 * c * Dd + t * 16 + Dd];   // + 4096 B
        og[t] = wmma4(pga, bv, og[t]);
        ol[t] = wmma4(pla, bv, ol[t]);
      }
    }
    __builtin_amdgcn_wave_barrier();

    kt += 16 * Dd;
    vt += 16 * Dd;
    at += 16;
    mt += 16;
  }

  // ---- epilogue: normalize, sigmoid-gate mix, signed sqrt ----
  const float gate = 1.0f / (1.0f + __expf(-alpha[0]));
  float* tp = tmp + (size_t)(b * Nn + qb) * Dd + h * HDm;
  const int toff = 8 * hi * Dd + col;   // + imm r*Dd + t*16
#pragma unroll
  for (int t = 0; t < 4; ++t) {
#pragma unroll
    for (int r = 0; r < 8; ++r) {
      const float vg = og[t][r] / lg[r];
      const float vl = ol[t][r] / ll[r];
      float x = gate * vg + (1.0f - gate) * vl;
      x = copysignf(sqrtf(fabsf(x)), x);
      tp[toff + r * Dd + t * 16] = x;
    }
  }
}

// ---------------------------------------------------------------------------
// L2 norm over axis=1 (sequence): rnorm[b,d] = 1 / max(||tmp[b,:,d]||, 1e-12)
// Coalesced: consecutive threads -> consecutive d.
// ---------------------------------------------------------------------------
__global__ void col_rnorm(const float* __restrict__ tmp, float* __restrict__ rnorm)
{
  const int idx = blockIdx.x * blockDim.x + threadIdx.x;   // 0 .. B*DV-1
  const int b = idx >> 10;
  const int d = idx & 1023;
  const float* p = tmp + (size_t)b * Nn * Dd + d;
  float acc = 0.0f;
  for (int n = 0; n < Nn; ++n) {
    const float v = p[(size_t)n * Dd];
    acc = fmaf(v, v, acc);
  }
  rnorm[idx] = 1.0f / fmaxf(sqrtf(acc), 1e-12f);
}

// ---------------------------------------------------------------------------
extern "C" void kernel_launch(void* const* d_in, const int* in_sizes, int n_in,
                              void* d_out, int out_size, void* d_ws, size_t ws_size,
                              hipStream_t stream)
{
  const float* query = (const float*)d_in[0];
  const float* key   = (const float*)d_in[1];
  const float* value = (const float*)d_in[2];
  const float* adj   = (const float*)d_in[3];
  const int*   mask  = (const int*)d_in[4];
  const float* Wq    = (const float*)d_in[5];
  const float* bq    = (const float*)d_in[6];
  const float* Wk    = (const float*)d_in[7];
  const float* bk    = (const float*)d_in[8];
  const float* Wv    = (const float*)d_in[9];
  const float* bv    = (const float*)d_in[10];
  const float* Wo    = (const float*)d_in[11];
  const float* bo    = (const float*)d_in[12];
  const float* alpha = (const float*)d_in[13];

  // workspace: Q | K | V | tmp (each B*N*D f32 = 16 MB) | rnorm (16 KB)
  float* Qp    = (float*)d_ws;
  float* Kp    = Qp  + (size_t)Bb * Nn * Dd;
  float* Vp    = Kp  + (size_t)Bb * Nn * Dd;
  float* tmp   = Vp  + (size_t)Bb * Nn * Dd;
  float* rnorm = tmp + (size_t)Bb * Nn * Dd;

  const dim3 gGrid(Bb * Nn / 64, Dd / 128);   // 64 x 8 workgroups, 128 thr
  gemm_xwT_wmma<false><<<gGrid, 128, 0, stream>>>(query, Wq, bq, nullptr, Qp, Dd, Dd, Nn);
  gemm_xwT_wmma<false><<<gGrid, 128, 0, stream>>>(key,   Wk, bk, nullptr, Kp, Dd, Dd, Nn);
  gemm_xwT_wmma<false><<<gGrid, 128, 0, stream>>>(value, Wv, bv, nullptr, Vp, Dd, Dd, Nn);

  const dim3 aGrid(Nn / 64, Hh, Bb);          // 16 x 16 x 4 workgroups
  attn_flash_dual<<<aGrid, 128, 0, stream>>>(Qp, Kp, Vp, adj, mask, alpha, tmp);

  col_rnorm<<<(Bb * Dd) / 256, 256, 0, stream>>>(tmp, rnorm);

  // output projection with per-(b,k) normalization folded into the A operand
  gemm_xwT_wmma<true><<<gGrid, 128, 0, stream>>>(tmp, Wo, bo, rnorm, (float*)d_out, Dd, Dd, Nn);
}